// KWinnersTakeAll_34084860461198
// MI455X (gfx1250) — compile-verified
//
#include <hip/hip_runtime.h>
#include <cstdint>

#define ROW_N    8192
#define THREADS  256
#define VEC_PER_T (ROW_N / 4 / THREADS)   // 8 uint4 per thread
#define KSEL     410                       // ceil(0.05 * 8192)

typedef __attribute__((ext_vector_type(4))) unsigned int v4u;
typedef __attribute__((ext_vector_type(8))) int          v8i;
typedef __attribute__((ext_vector_type(4))) int          v4i;

__device__ __forceinline__ unsigned relu_bits(unsigned b) {
    // keep bit pattern if float > 0, else 0 (monotone uint order for x >= 0)
    return (__uint_as_float(b) > 0.0f) ? b : 0u;
}

__global__ __launch_bounds__(THREADS)
void kwta_rowselect_kernel(const float* __restrict__ x, float* __restrict__ out)
{
    __shared__ unsigned rowbits[ROW_N];   // 32 KB row cache (relu'd bit patterns)
    __shared__ unsigned hist[256];        // per-pass digit histogram / suffix scan
    __shared__ unsigned s_prefix;
    __shared__ unsigned s_remaining;
    __shared__ unsigned s_tie;

    const unsigned tid      = threadIdx.x;
    const unsigned row      = blockIdx.x;
    const uint64_t row_base = (uint64_t)row * ROW_N;

    // ------------------------------------------------------------------
    // 1) Stage row HBM -> LDS via Tensor Data Mover (async DMA).
    //    One descriptor issued by wave 0; TDM ignores EXEC, issues per-wave.
    // ------------------------------------------------------------------
    const unsigned lds_off = (unsigned)(uintptr_t)(&rowbits[0]);
    if (tid < 32) {
        const uint64_t gaddr = (uint64_t)(uintptr_t)(x + row_base);
        v4u g0;
        g0.x = 1u;                                   // count=1, user descriptor
        g0.y = lds_off;                              // lds_addr (bytes)
        g0.z = (unsigned)(gaddr & 0xFFFFFFFFu);      // global_addr[31:0]
        g0.w = (unsigned)((gaddr >> 32) & 0x1FFFFFFu) | (2u << 30); // addr[56:32] | type=2

        v8i g1;
        g1[0] = 0x00020000;                    // workgroup_mask=0, data_size=2 (4 bytes)
        g1[1] = (int)((unsigned)ROW_N << 16);  // tensor_dim0[15:0] at bits[63:48]
        g1[2] = 0x00010000;                    // tensor_dim0 hi=0, tensor_dim1=1
        g1[3] = (int)((unsigned)ROW_N << 16);  // tile_dim0 at bits[127:112]
        g1[4] = 1;                             // tile_dim1=1, tile_dim2=0
        g1[5] = ROW_N;                         // tensor_dim0_stride[31:0]
        g1[6] = (int)((unsigned)ROW_N << 16);  // stride0 hi=0, tensor_dim1_stride lo16
        g1[7] = 0;                             // tensor_dim1_stride hi

        v4i g2 = {0, 0, 0, 0};                 // 2D tensor: groups 2/3 unused
        v4i g3 = {0, 0, 0, 0};
        v8i g4 = {0, 0, 0, 0, 0, 0, 0, 0};     // clang-23 6-arg form: extra group
        __builtin_amdgcn_tensor_load_to_lds(g0, g1, g2, g3, g4, 0);
        __builtin_amdgcn_s_wait_tensorcnt(0);
    }
    __syncthreads();

    // ------------------------------------------------------------------
    // 2) In-place relu -> orderable uint bits (LDS b128 traffic).
    // ------------------------------------------------------------------
    uint4* rb4 = (uint4*)rowbits;
#pragma unroll
    for (int j = 0; j < VEC_PER_T; ++j) {
        const unsigned i4 = tid + j * THREADS;
        uint4 v = rb4[i4];
        v.x = relu_bits(v.x); v.y = relu_bits(v.y);
        v.z = relu_bits(v.z); v.w = relu_bits(v.w);
        rb4[i4] = v;
    }
    __syncthreads();

    // ------------------------------------------------------------------
    // 3) Radix select (MSB-first, 4 x 8-bit digits) for the k-th largest.
    // ------------------------------------------------------------------
    unsigned prefix    = 0;
    unsigned remaining = KSEL;

#pragma unroll
    for (int pass = 0; pass < 4; ++pass) {
        const int      shift  = 24 - 8 * pass;
        const unsigned himask = (pass == 0) ? 0u : (0xFFFFFFFFu << (shift + 8));

        hist[tid] = 0;
        __syncthreads();

#pragma unroll
        for (int j = 0; j < VEC_PER_T; ++j) {
            const uint4 v = rb4[tid + j * THREADS];
            const unsigned e[4] = {v.x, v.y, v.z, v.w};
#pragma unroll
            for (int c = 0; c < 4; ++c) {
                if (((e[c] ^ prefix) & himask) == 0u)
                    atomicAdd(&hist[(e[c] >> shift) & 0xFFu], 1u);  // ds_add_u32
            }
        }
        __syncthreads();

        // Suffix scan: hist[d] := count of candidates with digit >= d.
#pragma unroll
        for (int off = 1; off < 256; off <<= 1) {
            const unsigned add = (tid + off < 256) ? hist[tid + off] : 0u;
            __syncthreads();
            hist[tid] += add;
            __syncthreads();
        }

        const unsigned cge      = hist[tid];
        const unsigned cge_next = (tid == 255) ? 0u : hist[tid + 1];
        if (cge >= remaining && cge_next < remaining) {   // exactly one thread
            s_prefix    = prefix | (tid << shift);
            s_remaining = remaining - cge_next;
        }
        __syncthreads();
        prefix    = s_prefix;
        remaining = s_remaining;
        __syncthreads();
    }

    const unsigned T = prefix;     // exact k-th largest of relu(x) (bit pattern)

    if (tid == 0) s_tie = 0;
    __syncthreads();

    // ------------------------------------------------------------------
    // 4) Emit binary mask, coalesced float4 stores (global_store_b128).
    //    T==0 => fewer than k positives: winners are exactly the strict
    //    positives (argmax included); all-nonpositive row keeps index 0.
    // ------------------------------------------------------------------
    float4* out4 = (float4*)(out + row_base);
#pragma unroll
    for (int j = 0; j < VEC_PER_T; ++j) {
        const unsigned i4 = tid + j * THREADS;
        const uint4 v = rb4[i4];
        const unsigned e[4] = {v.x, v.y, v.z, v.w};
        float m[4];
#pragma unroll
        for (int c = 0; c < 4; ++c) {
            if (e[c] > T) {
                m[c] = 1.0f;
            } else if (e[c] == T && T != 0u) {
                const unsigned slot = atomicAdd(&s_tie, 1u);  // ds_add_rtn_u32
                m[c] = (slot < remaining) ? 1.0f : 0.0f;
            } else {
                m[c] = 0.0f;
            }
        }
        out4[i4] = make_float4(m[0], m[1], m[2], m[3]);
    }

    // Degenerate row (h identically 0): reference keeps element 0.
    if (T == 0u && remaining == KSEL && tid == 0)
        out[row_base] = 1.0f;
}

extern "C" void kernel_launch(void* const* d_in, const int* in_sizes, int n_in,
                              void* d_out, int out_size, void* d_ws, size_t ws_size,
                              hipStream_t stream) {
    (void)n_in; (void)d_ws; (void)ws_size; (void)out_size;
    const float* x  = (const float*)d_in[0];
    float* out      = (float*)d_out;
    const int rows  = in_sizes[0] / ROW_N;   // 8192
    kwta_rowselect_kernel<<<rows, THREADS, 0, stream>>>(x, out);
}